// Rethinking_2731599200858
// MI455X (gfx1250) — compile-verified
//
#include <hip/hip_runtime.h>
#include <math.h>

// Problem sizes (compile-time constants from the reference)
#define VOCAB 16000
#define RANK  16
#define BB    64
#define TT    100
#define MROWS (BB * TT)      // 6400

typedef __attribute__((ext_vector_type(2))) float v2f;
typedef __attribute__((ext_vector_type(8))) float v8f;

// ---------------------------------------------------------------------------
// Kernel 1: qkv = x @ W_attn^T + b_attn
//   x:     (6400, 16000) row-major
//   W_attn:(48, 16000)   row-major
//   qkv:   (6400, 48)
// One block per 16-row M tile; 8 waves split K=16000 into chunks of 2000.
// Each wave: 3 accumulator tiles (N = 0..15, 16..31, 32..47), A fragment of x
// reused across the 3 B fragments. Deterministic LDS reduction across waves.
// ---------------------------------------------------------------------------
__global__ __launch_bounds__(256) void qkv_gemm_kernel(
    const float* __restrict__ x,
    const float* __restrict__ Wa,
    const float* __restrict__ ba,
    float* __restrict__ qkv)
{
    __shared__ float red[8 * 3 * 256];   // [wave][ntile][16x16]

    const int tid   = threadIdx.x;
    const int lane  = tid & 31;
    const int wave  = tid >> 5;
    const int m0    = blockIdx.x * 16;
    const int row   = lane & 15;         // M row (A) / N col (B,C)
    const int khalf = (lane >> 4) * 2;   // 0 for lanes 0-15, 2 for lanes 16-31

    const int kBeg = wave * (VOCAB / 8); // 2000 per wave
    const int kEnd = kBeg + (VOCAB / 8);

    v8f c0 = {}; v8f c1 = {}; v8f c2 = {};

    const float* xrow = x  + (size_t)(m0 + row) * VOCAB;
    const float* w0   = Wa + (size_t)(0  + row) * VOCAB;
    const float* w1   = Wa + (size_t)(16 + row) * VOCAB;
    const float* w2   = Wa + (size_t)(32 + row) * VOCAB;

    #pragma unroll 4
    for (int k = kBeg; k < kEnd; k += 4) {
        const int kk = k + khalf;
        v2f a  = *(const v2f*)(xrow + kk);
        v2f b0 = *(const v2f*)(w0 + kk);
        v2f b1 = *(const v2f*)(w1 + kk);
        v2f b2 = *(const v2f*)(w2 + kk);
        c0 = __builtin_amdgcn_wmma_f32_16x16x4_f32(false, a, false, b0, (short)0, c0, false, false);
        c1 = __builtin_amdgcn_wmma_f32_16x16x4_f32(false, a, false, b1, (short)0, c1, false, false);
        c2 = __builtin_amdgcn_wmma_f32_16x16x4_f32(false, a, false, b2, (short)0, c2, false, false);
    }

    // Dump per-wave partial tiles to LDS.
    const int mBase = (lane >> 4) * 8;
    #pragma unroll
    for (int i = 0; i < 8; ++i) {
        red[(wave * 3 + 0) * 256 + (mBase + i) * 16 + row] = c0[i];
        red[(wave * 3 + 1) * 256 + (mBase + i) * 16 + row] = c1[i];
        red[(wave * 3 + 2) * 256 + (mBase + i) * 16 + row] = c2[i];
    }
    __syncthreads();

    // Deterministic fixed-order reduction across the 8 waves, add bias, store.
    for (int e = tid; e < 16 * 48; e += 256) {
        const int m = e / 48, n = e % 48;
        const int t = n >> 4, n16 = n & 15;
        float s = 0.0f;
        #pragma unroll
        for (int w = 0; w < 8; ++w)
            s += red[(w * 3 + t) * 256 + m * 16 + n16];
        qkv[(size_t)(m0 + m) * 48 + n] = s + ba[n];
    }
}

// ---------------------------------------------------------------------------
// Kernel 2: rank-16 causal attention, one block per batch.
// Online (flash-style) softmax, one thread per query row. bias already folded
// into qkv by kernel 1.
// ---------------------------------------------------------------------------
__global__ __launch_bounds__(128) void attn_kernel(
    const float* __restrict__ qkv,
    float* __restrict__ y)
{
    __shared__ float qs[TT][17];
    __shared__ float ks[TT][17];
    __shared__ float vs[TT][17];

    const int b   = blockIdx.x;
    const int tid = threadIdx.x;

    for (int idx = tid; idx < TT * 48; idx += 128) {
        const int s = idx / 48, j = idx % 48;
        const float val = qkv[(size_t)(b * TT + s) * 48 + j];
        const int d = j & 15;
        if (j < 16)      qs[s][d] = val;
        else if (j < 32) ks[s][d] = val;
        else             vs[s][d] = val;
    }
    __syncthreads();

    const int t = tid;
    if (t < TT) {
        float mx = -INFINITY, l = 0.0f;
        float acc[RANK];
        #pragma unroll
        for (int d = 0; d < RANK; ++d) acc[d] = 0.0f;

        for (int s = 0; s <= t; ++s) {
            float dot = 0.0f;
            #pragma unroll
            for (int d = 0; d < RANK; ++d) dot += qs[t][d] * ks[s][d];
            const float sc  = dot * 0.25f;              // 1/sqrt(16)
            const float nm  = fmaxf(mx, sc);
            const float fac = __expf(mx - nm);          // exp(-inf)=0 on first iter
            const float p   = __expf(sc - nm);
            l = l * fac + p;
            #pragma unroll
            for (int d = 0; d < RANK; ++d) acc[d] = acc[d] * fac + p * vs[s][d];
            mx = nm;
        }
        const float inv = 1.0f / l;
        #pragma unroll
        for (int d = 0; d < RANK; ++d)
            y[(size_t)(b * TT + t) * RANK + d] = acc[d] * inv;
    }
}

// ---------------------------------------------------------------------------
// Kernel 3: out = gelu_exact(y @ W_proj^T + b_proj)
//   y:      (6400, 16)
//   W_proj: (16000, 16) row-major
//   out:    (6400, 16000)
// One 16x16 output tile per wave, K=16 -> 4 fp32 WMMAs. grid=(250,400), 4 waves.
// ---------------------------------------------------------------------------
__global__ __launch_bounds__(128) void proj_gelu_kernel(
    const float* __restrict__ y,
    const float* __restrict__ Wp,
    const float* __restrict__ bp,
    float* __restrict__ out)
{
    const int tid   = threadIdx.x;
    const int lane  = tid & 31;
    const int wave  = tid >> 5;
    const int m0    = blockIdx.y * 16;
    const int c0    = (blockIdx.x * 4 + wave) * 16;
    const int row   = lane & 15;
    const int khalf = (lane >> 4) * 2;

    const float* ya = y  + (size_t)(m0 + row) * RANK;
    const float* wb = Wp + (size_t)(c0 + row) * RANK;

    v8f c = {};
    #pragma unroll
    for (int k0 = 0; k0 < RANK; k0 += 4) {
        v2f a = *(const v2f*)(ya + k0 + khalf);
        v2f b = *(const v2f*)(wb + k0 + khalf);
        c = __builtin_amdgcn_wmma_f32_16x16x4_f32(false, a, false, b, (short)0, c, false, false);
    }

    const float bias  = bp[c0 + row];
    const int   mBase = m0 + (lane >> 4) * 8;
    #pragma unroll
    for (int i = 0; i < 8; ++i) {
        const float v = c[i] + bias;
        const float g = 0.5f * v * (1.0f + erff(v * 0.70710678118654752f)); // exact erf GELU
        out[(size_t)(mBase + i) * VOCAB + (c0 + row)] = g;
    }
}

// ---------------------------------------------------------------------------
extern "C" void kernel_launch(void* const* d_in, const int* in_sizes, int n_in,
                              void* d_out, int out_size, void* d_ws, size_t ws_size,
                              hipStream_t stream)
{
    const float* x  = (const float*)d_in[0];   // (64,100,16000)
    const float* Wa = (const float*)d_in[1];   // (48,16000)
    const float* ba = (const float*)d_in[2];   // (48,)
    const float* Wp = (const float*)d_in[3];   // (16000,16)
    const float* bp = (const float*)d_in[4];   // (16000,)
    float* out = (float*)d_out;                // (64,100,16000)

    float* qkv = (float*)d_ws;                 // 6400*48 floats = 1.23 MB
    float* yv  = qkv + (size_t)MROWS * 48;     // 6400*16 floats = 0.41 MB

    qkv_gemm_kernel<<<MROWS / 16, 256, 0, stream>>>(x, Wa, ba, qkv);
    attn_kernel<<<BB, 128, 0, stream>>>(qkv, yv);
    proj_gelu_kernel<<<dim3(VOCAB / 16 / 4, MROWS / 16), 128, 0, stream>>>(yv, Wp, bp, out);
}